// MaskNeighborSampler_47364899340959
// MI455X (gfx1250) — compile-verified
//
#include <hip/hip_runtime.h>
#include <stdint.h>

// Temporal neighbor sampler for MI455X (gfx1250, wave32).
// One wave per batch row, one sample per lane.
// CDNA5 paths used: global_load_async_to_lds_b64 (ASYNCcnt) staging,
// s_wait_asynccnt, wave32 ballot/popc for the valid-prefix count,
// LDS gathers for the random column samples.

#define MAX_DEG 64
#define WAVES_PER_BLOCK 8
#define BLOCK_THREADS (WAVES_PER_BLOCK * 32)

__global__ __launch_bounds__(BLOCK_THREADS)
void MaskNeighborSampler_kernel(const int* __restrict__ ids,
                                const float* __restrict__ tss,
                                const int* __restrict__ adj_info,
                                const float* __restrict__ ts_info,
                                const float* __restrict__ u,
                                float* __restrict__ out,  // [B*S] neighbors, then [B*S] ts
                                int B, int S)
{
    const int lane = threadIdx.x & 31;
    const int wave = threadIdx.x >> 5;
    const int row  = blockIdx.x * WAVES_PER_BLOCK + wave;
    if (row >= B) return;   // whole-wave uniform exit

    // Per-wave private LDS staging buffers (256B ts + 256B adj each).
    __shared__ __align__(16) float ts_lds [WAVES_PER_BLOCK][MAX_DEG];
    __shared__ __align__(16) int   adj_lds[WAVES_PER_BLOCK][MAX_DEG];

    const int   id = ids[row];
    const float ts = tss[row];
    const long long base = (long long)id * MAX_DEG;

    // ---- CDNA5 async global -> LDS staging (8B per lane per row) ----
    {
        uint64_t g_ts  = (uint64_t)(uintptr_t)(ts_info  + base) + (uint64_t)(lane * 8);
        uint64_t g_adj = (uint64_t)(uintptr_t)(adj_info + base) + (uint64_t)(lane * 8);
        // Generic LDS pointer: low 32 bits are the workgroup-relative LDS offset.
        uint32_t l_ts  = (uint32_t)(uintptr_t)(&ts_lds [wave][0]) + (uint32_t)(lane * 8);
        uint32_t l_adj = (uint32_t)(uintptr_t)(&adj_lds[wave][0]) + (uint32_t)(lane * 8);
        asm volatile("global_load_async_to_lds_b64 %0, %1, off"
                     :: "v"(l_ts),  "v"(g_ts)  : "memory");
        asm volatile("global_load_async_to_lds_b64 %0, %1, off"
                     :: "v"(l_adj), "v"(g_adj) : "memory");
        asm volatile("s_wait_asynccnt 0x0" ::: "memory");
    }
    // Wave-private LDS slice: no barrier needed.

    // ---- valid-prefix count via wave32 ballot (2 elements per lane) ----
    const float2 t2 = *(const float2*)&ts_lds[wave][lane * 2];
    unsigned long long b0 = __ballot(t2.x < ts);
    unsigned long long b1 = __ballot(t2.y < ts);
    const int count = __popcll(b0) + __popcll(b1);

    const int   err  = (count > 0) ? count : 1;
    const float ferr = (float)err;
    const int   hi   = err - 1;
    const int   zero = (count > 0) ? 1 : 0;

    const float* tsl = &ts_lds [wave][0];
    const int*   adl = &adj_lds[wave][0];
    float* outN = out;
    float* outT = out + (size_t)B * (size_t)S;

    // ---- bounded sampling: one sample per lane (loop covers S != 32) ----
    for (int s = lane; s < S; s += 32) {
        const size_t o  = (size_t)row * (size_t)S + (size_t)s;
        const float  uu = u[o];
        int col = (int)(uu * ferr);
        col = (col < hi) ? col : hi;

        const int   nb = zero * adl[col];
        const float tv = zero ? tsl[col] : 0.0f;

        outN[o] = (float)nb;   // neighbor ids < 2^24: exact in f32
        outT[o] = tv;
    }
}

extern "C" void kernel_launch(void* const* d_in, const int* in_sizes, int n_in,
                              void* d_out, int out_size, void* d_ws, size_t ws_size,
                              hipStream_t stream) {
    const int*   ids      = (const int*)  d_in[0];
    const float* tss      = (const float*)d_in[1];
    const int*   adj_info = (const int*)  d_in[2];
    const float* ts_info  = (const float*)d_in[3];
    const float* u        = (const float*)d_in[4];
    // d_in[5] = batch_size scalar, d_in[6] = num_samples scalar (device side);
    // derive the same values host-side from flat sizes.
    const int B = in_sizes[0];
    const int S = (B > 0) ? (in_sizes[4] / B) : 0;

    float* out = (float*)d_out;

    if (B <= 0 || S <= 0) return;
    const int grid = (B + WAVES_PER_BLOCK - 1) / WAVES_PER_BLOCK;
    MaskNeighborSampler_kernel<<<grid, BLOCK_THREADS, 0, stream>>>(
        ids, tss, adj_info, ts_info, u, out, B, S);
}